// plane_rotations_40243843564155
// MI455X (gfx1250) — compile-verified
//
#include <hip/hip_runtime.h>
#include <hip/hip_bf16.h>

#define NDIM   128
#define NROT   (NDIM * (NDIM - 1) / 2)        // 8128
#define CHUNK  32
#define NCHUNK ((NROT + CHUNK - 1) / CHUNK)   // 254

typedef __attribute__((ext_vector_type(2))) float v2f;
typedef __attribute__((ext_vector_type(8))) float v8f;

// ---------------------------------------------------------------------------
// Phase 1: build the 128x128 rotation matrix.
// The zeros-based plane-rotation scan reduces exactly to an ordered product of
// 2x2 matrices T_s = R(theta_s) * G_s applied to initial state [e0; e1], where
// G_s selects which of the two live rows feed rows i and j of step s.
// Parallel chunked scan: 254 chunks of 32 sequential steps, serial combine.
// ---------------------------------------------------------------------------
__device__ __forceinline__ void pair_at(int s, int* pi, int* pj) {
    int i = 0, off = 0;
    while (s >= off + (NDIM - 1 - i)) { off += NDIM - 1 - i; ++i; }
    *pi = i;
    *pj = i + 1 + (s - off);
}

__global__ __launch_bounds__(256) void build_mat_kernel(const float* __restrict__ rots,
                                                        float* __restrict__ mat) {
    __shared__ float cp[256][4];
    __shared__ float Psh[4];
    const int t = threadIdx.x;

    // identity by default (empty chunks)
    float p00 = 1.f, p01 = 0.f, p10 = 0.f, p11 = 1.f;

    const int s0 = t * CHUNK;
    if (s0 < NROT) {
        int send = s0 + CHUNK; if (send > NROT) send = NROT;
        int i, j, pi, pj;
        pair_at(s0, &i, &j);
        if (s0 == 0) { pi = 0; pj = 1; }           // virtual previous pair: state = [e0;e1]
        else         { pair_at(s0 - 1, &pi, &pj); }

        for (int s = s0; s < send; ++s) {
            const float ang = rots[s];
            const float c  = cosf(ang);
            const float sn = sinf(ang);
            const float g00 = (i == pi) ? 1.f : 0.f;
            const float g01 = (i == pj) ? 1.f : 0.f;
            const float g10 = (j == pi) ? 1.f : 0.f;
            const float g11 = (j == pj) ? 1.f : 0.f;
            // T = [[c,-sn],[sn,c]] * G
            const float t00 = c * g00 - sn * g10;
            const float t01 = c * g01 - sn * g11;
            const float t10 = sn * g00 + c * g10;
            const float t11 = sn * g01 + c * g11;
            // p = T * p (apply T after everything before it)
            const float n00 = t00 * p00 + t01 * p10;
            const float n01 = t00 * p01 + t01 * p11;
            const float n10 = t10 * p00 + t11 * p10;
            const float n11 = t10 * p01 + t11 * p11;
            p00 = n00; p01 = n01; p10 = n10; p11 = n11;
            pi = i; pj = j;
            if (++j == NDIM) { ++i; j = i + 1; }
        }
    }
    cp[t][0] = p00; cp[t][1] = p01; cp[t][2] = p10; cp[t][3] = p11;
    __syncthreads();

    if (t == 0) {
        float a00 = 1.f, a01 = 0.f, a10 = 0.f, a11 = 1.f;
        for (int u = 0; u < NCHUNK; ++u) {   // P = C_{last} * ... * C_0
            const float b00 = cp[u][0], b01 = cp[u][1], b10 = cp[u][2], b11 = cp[u][3];
            const float n00 = b00 * a00 + b01 * a10;
            const float n01 = b00 * a01 + b01 * a11;
            const float n10 = b10 * a00 + b11 * a10;
            const float n11 = b10 * a01 + b11 * a11;
            a00 = n00; a01 = n01; a10 = n10; a11 = n11;
        }
        Psh[0] = a00; Psh[1] = a01; Psh[2] = a10; Psh[3] = a11;
    }
    __syncthreads();

    const float P00 = Psh[0], P01 = Psh[1], P10 = Psh[2], P11 = Psh[3];
    // final live rows are the last step's (i,j) = (NDIM-2, NDIM-1); their
    // contents are combinations of e0,e1 with coefficients from P.
    for (int idx = t; idx < NDIM * NDIM; idx += 256) {
        const int r = idx >> 7;        // row
        const int c = idx & (NDIM - 1);
        float v = 0.f;
        if      (r == NDIM - 2) v = (c == 0) ? P00 : (c == 1) ? P01 : 0.f;
        else if (r == NDIM - 1) v = (c == 0) ? P10 : (c == 1) ? P11 : 0.f;
        mat[idx] = v;
    }
}

// ---------------------------------------------------------------------------
// Phase 2: out = x @ mat.T  via V_WMMA_F32_16X16X4_F32.
// Block = 256 threads (8 waves); each wave owns a 16-row strip of x and all
// 8 column tiles (128 cols / 16). One A fragment per k-step feeds 8 WMMAs.
// 32-bit A 16x4 layout: lane l holds M = l%16, VGPR v holds K = v + 2*(l>=16).
// 32-bit B 4x16 layout: lane l holds N = l%16, VGPR v holds K = v + 2*(l>=16).
// 32-bit C/D 16x16:     lane l holds N = l%16, VGPR v holds M = v + 8*(l>=16).
// ---------------------------------------------------------------------------
__global__ __launch_bounds__(256) void rot_apply_wmma(const float* __restrict__ x,
                                                      const float* __restrict__ mat,
                                                      float* __restrict__ out) {
    const int lane = threadIdx.x & 31;
    const int wave = threadIdx.x >> 5;
    const int l16  = lane & 15;
    const int half = lane >> 4;
    const long m_base = ((long)blockIdx.x * 8 + wave) * 16;

    const float* aptr = x   + (m_base + l16) * NDIM + 2 * half;   // A[M=l16][K=2*half + v]
    const float* bptr = mat + (long)l16 * NDIM + 2 * half;        // B[K][N=l16] = mat[n][k]

    v8f acc[8] = {};

    #pragma unroll 2
    for (int k0 = 0; k0 < NDIM; k0 += 4) {
        const v2f a = *(const v2f*)(aptr + k0);
        #pragma unroll
        for (int nt = 0; nt < 8; ++nt) {
            const v2f b = *(const v2f*)(bptr + nt * (16 * NDIM) + k0);
            acc[nt] = __builtin_amdgcn_wmma_f32_16x16x4_f32(
                /*neg_a=*/false, a, /*neg_b=*/false, b,
                /*c_mod=*/(short)0, acc[nt], /*reuse_a=*/false, /*reuse_b=*/false);
        }
    }

    float* obase = out + m_base * NDIM + l16;
    #pragma unroll
    for (int nt = 0; nt < 8; ++nt) {
        #pragma unroll
        for (int v = 0; v < 8; ++v) {
            obase[(long)(v + 8 * half) * NDIM + nt * 16] = acc[nt][v];
        }
    }
}

extern "C" void kernel_launch(void* const* d_in, const int* in_sizes, int n_in,
                              void* d_out, int out_size, void* d_ws, size_t ws_size,
                              hipStream_t stream) {
    const float* x    = (const float*)d_in[0];
    const float* rots = (const float*)d_in[1];
    float* out = (float*)d_out;
    float* mat = (float*)d_ws;   // 128*128*4 = 64 KB scratch

    const int nrows = in_sizes[0] / NDIM;      // 262144

    build_mat_kernel<<<1, 256, 0, stream>>>(rots, mat);

    const int blocks = nrows / 128;            // 8 waves * 16 rows per block
    rot_apply_wmma<<<blocks, 256, 0, stream>>>(x, mat, out);
}